// Attention_29119878267652
// MI455X (gfx1250) — compile-verified
//
#include <hip/hip_runtime.h>
#include <stdint.h>
#include <math.h>

// B=4, S=2048, E=1024, H=16, D=64
#define SB 4
#define SS 2048
#define SE 1024
#define SH 16
#define SD 64

typedef float v2f __attribute__((ext_vector_type(2)));
typedef float v8f __attribute__((ext_vector_type(8)));
typedef uint32_t u32x4 __attribute__((ext_vector_type(4)));
typedef uint32_t u32x8 __attribute__((ext_vector_type(8)));

// D(16x16,f32) = A(16x4,f32) x B(4x16,f32) + C
// A layout: lane l -> M = l&15, VGPR v -> K = 2*(l>>4) + v  (float2 per lane)
// B layout (symmetric): lane l -> N = l&15, VGPR v -> K = 2*(l>>4) + v
// C/D layout: lane l -> N = l&15, VGPR v -> M = v + 8*(l>>4)
__device__ __forceinline__ v8f wmma4(v2f a, v2f b, v8f c) {
  return __builtin_amdgcn_wmma_f32_16x16x4_f32(false, a, false, b, (short)0, c,
                                               false, false);
}

// ISA 10.2: generic LDS pointer low 32 bits == wave-relative LDS byte address
__device__ __forceinline__ uint32_t lds_off(const void* p) {
  return (uint32_t)(uintptr_t)p;
}

// CDNA5 async DMA: memory -> LDS, 16B per lane, tracked by ASYNCcnt
__device__ __forceinline__ void async_copy16(uint32_t lds, const float* g) {
  asm volatile("global_load_async_to_lds_b128 %0, %1, off" ::"v"(lds),
               "v"((uint64_t)(uintptr_t)g)
               : "memory");
}
__device__ __forceinline__ void wait_async() {
  asm volatile("s_wait_asynccnt 0x0" ::: "memory");
}

// CDNA5 Tensor Data Mover: one-instruction 2D tile DMA (ISA ch.8).
// Copies a 64x64 f32 tile (row stride 3072 elems) into LDS, inserting
// pad_code+1 dwords of LDS padding after every 64 dwords (D# pad fields).
__device__ __forceinline__ void tdm_load_64x64(uint32_t lds, const float* g,
                                               uint32_t pad_code) {
  const uint64_t ga = (uint64_t)(uintptr_t)g;
  u32x4 g0;
  g0[0] = 1u;             // count=1, user mode, no gather
  g0[1] = lds;            // lds_addr [63:32]
  g0[2] = (uint32_t)ga;   // global_addr low  [95:64]
  g0[3] = (uint32_t)((ga >> 32) & 0x01FFFFFFu) | (2u << 30);  // addr hi | type=2
  u32x8 g1;
  g1[0] = (2u << 16)            // data_size = 4B
          | (1u << 20)          // pad_enable
          | (5u << 22)          // pad_interval: 2^(5+1)=64 dwords
          | (pad_code << 25);   // pad_amount: pad_code+1 dwords
  g1[1] = (64u << 16);          // tensor_dim0 = 64   [79:48]
  g1[2] = (64u << 16);          // tensor_dim1 = 64   [111:80]
  g1[3] = (64u << 16);          // tile_dim0   = 64   [127:112]
  g1[4] = 64u;                  // tile_dim1 = 64, tile_dim2 = 0
  g1[5] = 3072u;                // tensor_dim0_stride = 3E (elements)
  g1[6] = 0u;                   // stride0 hi, stride1 lo
  g1[7] = 0u;                   // stride1 hi
  asm volatile("tensor_load_to_lds %0, %1" ::"s"(g0), "s"(g1) : "memory");
}

// ---------------------------------------------------------------------------
// C[M,N] = A[M,K] @ B[K,N] + bias[N]   (row major, M%128==0, N%64==0, K%16==0)
// block tile 128x64, 8 waves, each wave 32x32; async double-buffered staging
// ---------------------------------------------------------------------------
__global__ __launch_bounds__(256) void gemm_bias_wmma(
    const float* __restrict__ A, const float* __restrict__ B,
    const float* __restrict__ bias, float* __restrict__ C, int M, int N,
    int K) {
  __shared__ float As[2][128][20];  // [m][k]; stride 80B: 16B-mult, no conflicts
  __shared__ float Bs[2][16][72];   // [k][n]; stride 288B: half-waves disjoint

  const int tid  = threadIdx.x;
  const int lane = tid & 31;
  const int wv   = tid >> 5;  // 0..7
  const int hi   = lane >> 4;
  const int ln   = lane & 15;
  const int wm   = (wv & 3) * 32;
  const int wn   = (wv >> 2) * 32;
  const long bm0 = (long)blockIdx.x * 128;
  const long bn0 = (long)blockIdx.y * 64;

  // staging coords (16B chunks)
  const int ar  = tid >> 2;        // A row (0..63), +64 for 2nd chunk
  const int ac  = (tid & 3) * 4;   // A col (floats)
  const int brr = tid >> 4;        // B row (0..15)
  const int bcc = (tid & 15) * 4;  // B col (floats)

  v8f acc[2][2];
#pragma unroll
  for (int i = 0; i < 2; ++i)
#pragma unroll
    for (int j = 0; j < 2; ++j) acc[i][j] = {};

  auto stage = [&](int t, int buf) {
    const int kb = t * 16;
    async_copy16(lds_off(&As[buf][ar][ac]),
                 &A[(bm0 + ar) * (long)K + kb + ac]);
    async_copy16(lds_off(&As[buf][ar + 64][ac]),
                 &A[(bm0 + ar + 64) * (long)K + kb + ac]);
    async_copy16(lds_off(&Bs[buf][brr][bcc]),
                 &B[(long)(kb + brr) * N + bn0 + bcc]);
  };

  const int nkb = K >> 4;
  stage(0, 0);
  for (int t = 0; t < nkb; ++t) {
    const int cur = t & 1;
    wait_async();     // my async writes for buf[cur] landed
    __syncthreads();  // everyone's landed; everyone done reading buf[cur^1]
    if (t + 1 < nkb) stage(t + 1, cur ^ 1);

#pragma unroll
    for (int ks = 0; ks < 4; ++ks) {
      const int kk = ks * 4 + hi * 2;
      const v2f a0 = *reinterpret_cast<const v2f*>(&As[cur][wm + ln][kk]);
      const v2f a1 = *reinterpret_cast<const v2f*>(&As[cur][wm + 16 + ln][kk]);
      v2f b0, b1;
      b0[0] = Bs[cur][kk + 0][wn + ln];
      b0[1] = Bs[cur][kk + 1][wn + ln];
      b1[0] = Bs[cur][kk + 0][wn + 16 + ln];
      b1[1] = Bs[cur][kk + 1][wn + 16 + ln];
      acc[0][0] = wmma4(a0, b0, acc[0][0]);
      acc[0][1] = wmma4(a0, b1, acc[0][1]);
      acc[1][0] = wmma4(a1, b0, acc[1][0]);
      acc[1][1] = wmma4(a1, b1, acc[1][1]);
    }
  }

#pragma unroll
  for (int i = 0; i < 2; ++i)
#pragma unroll
    for (int j = 0; j < 2; ++j) {
      const long col = bn0 + wn + j * 16 + ln;
      const float bv = bias[col];
#pragma unroll
      for (int v = 0; v < 8; ++v) {
        const long row = bm0 + wm + i * 16 + hi * 8 + v;
        C[row * (long)N + col] = acc[i][j][v] + bv;
      }
    }
}

// ---------------------------------------------------------------------------
// Flash attention over QKV buffer [B,S,3E]; out [B,S,E] (heads merged).
// Block = (64-row q-tile, head, batch); 4 waves x 16 q-rows.
// K/V tiles double-buffered via TDM (wave0: K, wave1: V), Q via async DMA.
// ---------------------------------------------------------------------------
__global__ __launch_bounds__(128) void flash_attn_wmma(
    const float* __restrict__ qkv, const float* __restrict__ amask,
    float* __restrict__ out) {
  __shared__ float Qs[64][68];     // [q][d]      float2 A-like reads
  __shared__ float Ks[2][64][68];  // [key][d]    float2 B-like reads (QK^T)
  __shared__ float Vs[2][64][72];  // [key][d]    scalar B-like reads (PV)
  __shared__ float Ps[4][16][68];  // per-wave P: [q][key]

  const int tid  = threadIdx.x;
  const int lane = tid & 31;
  const int wv   = tid >> 5;  // 0..3
  const int hi   = lane >> 4;
  const int ln   = lane & 15;
  const int qb   = blockIdx.x;  // 0..31
  const int h    = blockIdx.y;  // 0..15
  const int b    = blockIdx.z;  // 0..3
  const int q0   = qb * 64;
  const long LD  = 3 * SE;

  const float* qp = qkv + (long)b * SS * LD + (long)h * SD;
  const float* kp = qp + SE;
  const float* vp = qp + 2 * SE;

  const int sr = tid >> 4;        // staging row base (0..7), row = sr + 8*i
  const int sc = (tid & 15) * 4;  // staging col (floats)

  // One TDM descriptor per tile: wave0 moves K, wave1 moves V.
  auto stage_kv = [&](int j, int buf) {
    if (wv == 0)
      tdm_load_64x64(lds_off(&Ks[buf][0][0]), kp + (long)(j * 64) * LD, 3u);
    else if (wv == 1)
      tdm_load_64x64(lds_off(&Vs[buf][0][0]), vp + (long)(j * 64) * LD, 7u);
  };

  // stage Q tile (per-lane async) + first K/V tile (TDM)
#pragma unroll
  for (int i = 0; i < 8; ++i) {
    const int r = sr + i * 8;
    async_copy16(lds_off(&Qs[r][sc]), &qp[(long)(q0 + r) * LD + sc]);
  }
  stage_kv(0, 0);

  v2f qa[16];
  float mrow[8], lrow[8];
#pragma unroll
  for (int v = 0; v < 8; ++v) {
    mrow[v] = -1e30f;
    lrow[v] = 0.0f;
  }
  v8f o[4];
#pragma unroll
  for (int nt = 0; nt < 4; ++nt) o[nt] = {};

  for (int j = 0; j <= qb; ++j) {  // causal: skip blocks above diagonal
    const int cur = j & 1;
    if (wv < 2) __builtin_amdgcn_s_wait_tensorcnt(0);  // my TDM tile landed
    if (j == 0) wait_async();                          // Q landed
    __syncthreads();  // all tiles visible; buf[cur^1] free for reuse
    if (j == 0) {     // pull Q A-fragments once
#pragma unroll
      for (int ks = 0; ks < 16; ++ks)
        qa[ks] =
            *reinterpret_cast<const v2f*>(&Qs[wv * 16 + ln][ks * 4 + hi * 2]);
    }
    if (j < qb) stage_kv(j + 1, cur ^ 1);

    // S = Q @ K^T (row-major K tile is already B-fragment friendly)
    v8f s[4];
#pragma unroll
    for (int nt = 0; nt < 4; ++nt) s[nt] = {};
#pragma unroll
    for (int ks = 0; ks < 16; ++ks) {
      const int kk = ks * 4 + hi * 2;
#pragma unroll
      for (int nt = 0; nt < 4; ++nt) {
        const v2f bk =
            *reinterpret_cast<const v2f*>(&Ks[cur][nt * 16 + ln][kk]);
        s[nt] = wmma4(qa[ks], bk, s[nt]);
      }
    }

    float am[4];
#pragma unroll
    for (int nt = 0; nt < 4; ++nt)
      am[nt] = amask[(long)b * SS + j * 64 + nt * 16 + ln];

    // online softmax; row v+8*hi lives in this lane's 16-lane group
#pragma unroll
    for (int v = 0; v < 8; ++v) {
      const int qg = q0 + wv * 16 + hi * 8 + v;
      float rm = -1e30f;
#pragma unroll
      for (int nt = 0; nt < 4; ++nt) {
        const int kg = j * 64 + nt * 16 + ln;
        float sv = (kg <= qg) ? s[nt][v] * 0.125f : -10000.0f;  // 1/sqrt(64)
        sv += am[nt];
        s[nt][v] = sv;
        rm = fmaxf(rm, sv);
      }
      rm = fmaxf(rm, __shfl_xor(rm, 1, 16));
      rm = fmaxf(rm, __shfl_xor(rm, 2, 16));
      rm = fmaxf(rm, __shfl_xor(rm, 4, 16));
      rm = fmaxf(rm, __shfl_xor(rm, 8, 16));
      const float mn    = fmaxf(mrow[v], rm);
      const float alpha = __expf(mrow[v] - mn);
      mrow[v] = mn;
      float rs = 0.0f;
#pragma unroll
      for (int nt = 0; nt < 4; ++nt) {
        const float p = __expf(s[nt][v] - mn);
        s[nt][v] = p;
        rs += p;
      }
      rs += __shfl_xor(rs, 1, 16);
      rs += __shfl_xor(rs, 2, 16);
      rs += __shfl_xor(rs, 4, 16);
      rs += __shfl_xor(rs, 8, 16);
      lrow[v] = lrow[v] * alpha + rs;
#pragma unroll
      for (int nt = 0; nt < 4; ++nt) o[nt][v] *= alpha;
    }

    // P: C-layout -> LDS -> A-layout
#pragma unroll
    for (int nt = 0; nt < 4; ++nt)
#pragma unroll
      for (int v = 0; v < 8; ++v) Ps[wv][hi * 8 + v][nt * 16 + ln] = s[nt][v];
    __syncthreads();

    // O += P @ V (row-major V tile: B-fragment = 2 scalar reads)
#pragma unroll
    for (int ks = 0; ks < 16; ++ks) {
      const int kk = ks * 4 + hi * 2;
      const v2f pa = *reinterpret_cast<const v2f*>(&Ps[wv][ln][kk]);
#pragma unroll
      for (int nt = 0; nt < 4; ++nt) {
        v2f bv;
        bv[0] = Vs[cur][kk + 0][nt * 16 + ln];
        bv[1] = Vs[cur][kk + 1][nt * 16 + ln];
        o[nt] = wmma4(pa, bv, o[nt]);
      }
    }
  }

  // normalize + write merged-head output [B,S,E]
#pragma unroll
  for (int nt = 0; nt < 4; ++nt)
#pragma unroll
    for (int v = 0; v < 8; ++v) {
      const int row = q0 + wv * 16 + hi * 8 + v;
      out[((long)b * SS + row) * SE + h * SD + nt * 16 + ln] =
          o[nt][v] / lrow[v];
    }
}

// ---------------------------------------------------------------------------
extern "C" void kernel_launch(void* const* d_in, const int* in_sizes, int n_in,
                              void* d_out, int out_size, void* d_ws,
                              size_t ws_size, hipStream_t stream) {
  (void)in_sizes;
  (void)n_in;
  (void)out_size;
  (void)ws_size;
  const float* x     = (const float*)d_in[0];  // [4,2048,1024]
  const float* amask = (const float*)d_in[1];  // [4,1,1,2048]
  const float* Wa    = (const float*)d_in[2];  // [1024,3072]
  const float* ba    = (const float*)d_in[3];  // [3072]
  const float* Wp    = (const float*)d_in[4];  // [1024,1024]
  const float* bp    = (const float*)d_in[5];  // [1024]
  float* out = (float*)d_out;                  // [4,2048,1024]

  float* qkv = (float*)d_ws;                  // [8192,3072]
  float* att = qkv + (long)SB * SS * 3 * SE;  // [8192,1024]

  const int M = SB * SS;  // 8192
  {
    dim3 grid(M / 128, (3 * SE) / 64);
    gemm_bias_wmma<<<grid, 256, 0, stream>>>(x, Wa, ba, qkv, M, 3 * SE, SE);
  }
  {
    dim3 grid(SS / 64, SH, SB);
    flash_attn_wmma<<<grid, 128, 0, stream>>>(qkv, amask, att);
  }
  {
    dim3 grid(M / 128, SE / 64);
    gemm_bias_wmma<<<grid, 256, 0, stream>>>(att, Wp, bp, out, M, SE, SE);
  }
}